// AltAttention_68582037782991
// MI455X (gfx1250) — compile-verified
//
#include <hip/hip_runtime.h>
#include <hip/hip_bf16.h>

typedef float v2f __attribute__((ext_vector_type(2)));
typedef float v8f __attribute__((ext_vector_type(8)));

#define B_  2
#define S_  2048
#define D_  1024
#define H_  16
#define DH_ 64

// D(16x16,f32) = A(16x4,f32) * B(4x16,f32) + C   -- native fp32 WMMA on gfx1250
__device__ __forceinline__ v8f wmma4(v2f a, v2f b, v8f c) {
  return __builtin_amdgcn_wmma_f32_16x16x4_f32(
      /*neg_a=*/false, a, /*neg_b=*/false, b,
      /*c_mod=*/(short)0, c, /*reuse_a=*/false, /*reuse_b=*/false);
}

// ---------------------------------------------------------------------------
// Kernel 1: qkv = inputs @ Wqkv^T + bqkv, scattered to Q/K/V [B,H,S,dh]
// wave tile: 64 (M) x 64 (N); block = 4 waves covering 4 consecutive N tiles
// ---------------------------------------------------------------------------
__global__ __launch_bounds__(128) void qkv_gemm_kernel(
    const float* __restrict__ x, const float* __restrict__ W,
    const float* __restrict__ bias,
    float* __restrict__ qws, float* __restrict__ kws, float* __restrict__ vws)
{
  const int lane = threadIdx.x & 31;
  const int wave = threadIdx.x >> 5;
  const int lm = lane & 15, hi = lane >> 4;
  const int row0 = blockIdx.x * 64;
  const int col0 = (blockIdx.y * 4 + wave) * 64;

  v8f acc[4][4];
#pragma unroll
  for (int i = 0; i < 4; ++i)
#pragma unroll
    for (int j = 0; j < 4; ++j) acc[i][j] = {};

  const float* aBase = x + (size_t)(row0 + lm) * D_ + 2 * hi;
  const float* bBase = W + (size_t)(col0 + lm) * D_ + 2 * hi;

  for (int k0 = 0; k0 < D_; k0 += 4) {
    v2f a[4], b[4];
#pragma unroll
    for (int ms = 0; ms < 4; ++ms)
      a[ms] = *(const v2f*)(aBase + (size_t)ms * 16 * D_ + k0);
#pragma unroll
    for (int ns = 0; ns < 4; ++ns)
      b[ns] = *(const v2f*)(bBase + (size_t)ns * 16 * D_ + k0);
#pragma unroll
    for (int ms = 0; ms < 4; ++ms)
#pragma unroll
      for (int ns = 0; ns < 4; ++ns)
        acc[ms][ns] = wmma4(a[ms], b[ns], acc[ms][ns]);
  }

  // epilogue: bias + scatter to Q/K/V head-major layout
#pragma unroll
  for (int ns = 0; ns < 4; ++ns) {
    const int c0 = col0 + ns * 16;           // 16-aligned; 64-tile aligns with q/k/v segment
    const int h = c0 / (3 * DH_);
    const int rem = c0 % (3 * DH_);
    const int sel = rem / DH_;
    const int d0 = rem % DH_;
    float* dst = (sel == 0) ? qws : (sel == 1) ? kws : vws;
    const float bv = bias[c0 + lm];
#pragma unroll
    for (int ms = 0; ms < 4; ++ms) {
#pragma unroll
      for (int v = 0; v < 8; ++v) {
        const int r = row0 + ms * 16 + v + 8 * hi;   // global (b,s) row
        const int bb = r >> 11;
        const int ss = r & (S_ - 1);
        dst[((size_t)(bb * H_ + h) * S_ + ss) * DH_ + d0 + lm] = acc[ms][ns][v] + bv;
      }
    }
  }
}

// ---------------------------------------------------------------------------
// Kernel 2: flash attention per (b,h,16-row q block), one wave per q block
// ---------------------------------------------------------------------------
__global__ __launch_bounds__(128) void attn_kernel(
    const float* __restrict__ qws, const float* __restrict__ kws,
    const float* __restrict__ vws, const unsigned char* __restrict__ mask,
    float* __restrict__ xws)
{
  __shared__ float plds[4 * 256];                 // 1KB P-staging slot per wave
  const int lane = threadIdx.x & 31;
  const int wave = threadIdx.x >> 5;
  const int lm = lane & 15, hi = lane >> 4;
  const int b = blockIdx.z, h = blockIdx.y;
  const int q0 = (blockIdx.x * 4 + wave) * 16;

  const size_t headOff = (size_t)(b * H_ + h) * S_ * DH_;
  const float* Q = qws + headOff;
  const float* K = kws + headOff;
  const float* V = vws + headOff;

  // Q block resident in A-fragment layout: m=lm, k = 4f + v + 2*hi
  v2f qa[16];
#pragma unroll
  for (int f = 0; f < 16; ++f)
    qa[f] = *(const v2f*)(Q + (size_t)(q0 + lm) * DH_ + 4 * f + 2 * hi);

  v8f acc[4];
#pragma unroll
  for (int t = 0; t < 4; ++t) acc[t] = {};
  float mrow[8], lrow[8];
#pragma unroll
  for (int v = 0; v < 8; ++v) { mrow[v] = -3.0e38f; lrow[v] = 0.0f; }

  const float scale = 0.03125f;                       // D^-0.5 = 1024^-0.5
  const float slope = exp2f(-0.5f * (float)(h + 1));  // ALiBi slope per head
  float* slot = plds + wave * 256;

  for (int j0 = 0; j0 < S_; j0 += 16) {
    // S tile = Q * K^T  (B frag: n=lm -> key j0+lm, k along d)
    v8f s = {};
#pragma unroll
    for (int f = 0; f < 16; ++f) {
      v2f kb = *(const v2f*)(K + (size_t)(j0 + lm) * DH_ + 4 * f + 2 * hi);
      s = wmma4(qa[f], kb, s);
    }

    const int j = j0 + lm;                    // this lane's key column
    const bool valid = mask[(size_t)b * S_ + j] != 0;
    float sv[8];
#pragma unroll
    for (int v = 0; v < 8; ++v) {
      const int qi = q0 + v + 8 * hi;         // this element's query row
      const float val = s[v] * scale - slope * fabsf((float)(qi - j));
      sv[v] = valid ? val : -3.0e38f;
    }

    // row max across the 16 lanes holding one row (n dimension)
    float rmax[8];
#pragma unroll
    for (int v = 0; v < 8; ++v) {
      rmax[v] = sv[v];
#pragma unroll
      for (int off = 8; off >= 1; off >>= 1)
        rmax[v] = fmaxf(rmax[v], __shfl_xor(rmax[v], off, 32));
    }

    float alpha[8], p[8], rs[8];
#pragma unroll
    for (int v = 0; v < 8; ++v) {
      const float mn = fmaxf(mrow[v], rmax[v]);
      alpha[v] = __expf(mrow[v] - mn);
      mrow[v] = mn;
      p[v] = __expf(sv[v] - mn);
      rs[v] = p[v];
    }
#pragma unroll
    for (int v = 0; v < 8; ++v) {
#pragma unroll
      for (int off = 8; off >= 1; off >>= 1)
        rs[v] += __shfl_xor(rs[v], off, 32);
      lrow[v] = lrow[v] * alpha[v] + rs[v];
    }
#pragma unroll
    for (int t = 0; t < 4; ++t)
#pragma unroll
      for (int v = 0; v < 8; ++v) acc[t][v] *= alpha[v];

    // stage P through LDS to convert C-layout -> A-layout (same-wave slot)
#pragma unroll
    for (int v = 0; v < 8; ++v)
      slot[(v + 8 * hi) * 16 + lm] = p[v];
    asm volatile("s_wait_dscnt 0" ::: "memory");

    // X += P * V
#pragma unroll
    for (int dt = 0; dt < 4; ++dt) {
#pragma unroll
      for (int jc = 0; jc < 4; ++jc) {
        v2f pa = *(const v2f*)(slot + lm * 16 + 4 * jc + 2 * hi);
        const float* vp = V + (size_t)(j0 + 4 * jc + 2 * hi) * DH_ + dt * 16 + lm;
        v2f vb;
        vb.x = vp[0];
        vb.y = vp[DH_];
        acc[dt] = wmma4(pa, vb, acc[dt]);
      }
    }
    asm volatile("s_wait_dscnt 0" ::: "memory");  // WAR: reads done before next-tile stores
  }

  // normalize + write x[b, s, h*64 + d] (head transpose folded)
#pragma unroll
  for (int dt = 0; dt < 4; ++dt) {
#pragma unroll
    for (int v = 0; v < 8; ++v) {
      const int qi = q0 + v + 8 * hi;
      xws[((size_t)b * S_ + qi) * D_ + h * DH_ + dt * 16 + lm] = acc[dt][v] / lrow[v];
    }
  }
}

// ---------------------------------------------------------------------------
// Kernel 3: out = x @ Wproj^T + bproj
// ---------------------------------------------------------------------------
__global__ __launch_bounds__(128) void proj_gemm_kernel(
    const float* __restrict__ x, const float* __restrict__ W,
    const float* __restrict__ bias, float* __restrict__ out)
{
  const int lane = threadIdx.x & 31;
  const int wave = threadIdx.x >> 5;
  const int lm = lane & 15, hi = lane >> 4;
  const int row0 = blockIdx.x * 64;
  const int col0 = (blockIdx.y * 4 + wave) * 64;

  v8f acc[4][4];
#pragma unroll
  for (int i = 0; i < 4; ++i)
#pragma unroll
    for (int j = 0; j < 4; ++j) acc[i][j] = {};

  const float* aBase = x + (size_t)(row0 + lm) * D_ + 2 * hi;
  const float* bBase = W + (size_t)(col0 + lm) * D_ + 2 * hi;

  for (int k0 = 0; k0 < D_; k0 += 4) {
    v2f a[4], b[4];
#pragma unroll
    for (int ms = 0; ms < 4; ++ms)
      a[ms] = *(const v2f*)(aBase + (size_t)ms * 16 * D_ + k0);
#pragma unroll
    for (int ns = 0; ns < 4; ++ns)
      b[ns] = *(const v2f*)(bBase + (size_t)ns * 16 * D_ + k0);
#pragma unroll
    for (int ms = 0; ms < 4; ++ms)
#pragma unroll
      for (int ns = 0; ns < 4; ++ns)
        acc[ms][ns] = wmma4(a[ms], b[ns], acc[ms][ns]);
  }

#pragma unroll
  for (int ns = 0; ns < 4; ++ns) {
    const int c0 = col0 + ns * 16;
    const float bv = bias[c0 + lm];
#pragma unroll
    for (int ms = 0; ms < 4; ++ms) {
#pragma unroll
      for (int v = 0; v < 8; ++v) {
        const int r = row0 + ms * 16 + v + 8 * hi;
        out[(size_t)r * D_ + c0 + lm] = acc[ms][ns][v] + bv;
      }
    }
  }
}

// ---------------------------------------------------------------------------
extern "C" void kernel_launch(void* const* d_in, const int* in_sizes, int n_in,
                              void* d_out, int out_size, void* d_ws, size_t ws_size,
                              hipStream_t stream) {
  const float* inputs = (const float*)d_in[0];
  const unsigned char* mask = (const unsigned char*)d_in[1];
  const float* Wqkv = (const float*)d_in[2];
  const float* bqkv = (const float*)d_in[3];
  const float* Wproj = (const float*)d_in[4];
  const float* bproj = (const float*)d_in[5];
  float* out = (float*)d_out;

  float* ws = (float*)d_ws;
  const size_t headElems = (size_t)B_ * H_ * S_ * DH_;  // 4M floats each
  float* qws = ws;
  float* kws = ws + headElems;
  float* vws = ws + 2 * headElems;
  float* xws = ws + 3 * headElems;                      // [B,S,D]

  dim3 blk(128);
  // M=4096 -> 64 tiles of 64; N=3072 -> 48 tiles of 64 -> grid.y=12 (4 waves/block)
  qkv_gemm_kernel<<<dim3(64, 12), blk, 0, stream>>>(inputs, Wqkv, bqkv, qws, kws, vws);
  // S/16 = 128 q blocks per (b,h), 4 waves/block -> grid.x=32
  attn_kernel<<<dim3(32, H_, B_), blk, 0, stream>>>(qws, kws, vws, mask, xws);
  // N=1024 -> 16 tiles of 64 -> grid.y=4
  proj_gemm_kernel<<<dim3(64, 4), blk, 0, stream>>>(xws, Wproj, bproj, out);
}